// EcliphraPhotoField_56650618634582
// MI455X (gfx1250) — compile-verified
//
#include <hip/hip_runtime.h>
#include <cmath>

// ---------------------------------------------------------------------------
// EcliphraPhotoField for MI455X (gfx1250).
// Memory-bound elementwise + sparse scatter + global norm rescale.
// CDNA5 paths used: global_load_async_to_lds_b128 (ASYNCcnt-tracked async
// copy, double buffered, th:TH_LOAD_NT) + s_wait_asynccnt, nontemporal
// final stores. No matrix ops exist in this operator, so WMMA is not used.
// ---------------------------------------------------------------------------

#define H_DIM 4096
#define W_DIM 4096
#define K_ATT 256
#define RMAX_C 10
#define SIDE_C (2 * RMAX_C + 1)              /* 21 */
#define NPIX ((size_t)H_DIM * (size_t)W_DIM) /* 16,777,216 */

#define TPB 256
#define ITER 4
#define ELEMS_PER_BLOCK (TPB * 4 * ITER)     /* 4096 floats per block */
#define NBLK ((int)(NPIX / ELEMS_PER_BLOCK)) /* 4096 blocks */
#define SCALE_SLOT (2 * NBLK)                /* ws float index of scale */

typedef float v4f __attribute__((ext_vector_type(4)));  // native vector type

// Low 32 bits of a flat shared-aperture pointer == DS byte address
// (CDNA5 ISA 10.2: LDS address is the low 32 bits of the flat address; shared aperture low bits = 0).
__device__ __forceinline__ unsigned lds_b32_addr(const void* p) {
  return (unsigned)(size_t)p;
}

// ---------------------------------------------------------------------------
// 1) Scatter: one block per attractor, 21x21 masked Gaussian patch into delta.
// ---------------------------------------------------------------------------
__global__ __launch_bounds__(TPB) void scatter_kernel(
    const int* __restrict__ pos, const float* __restrict__ strength,
    const float* __restrict__ g_lratio, float* delta) {
  const int k = blockIdx.x;
  const float s = strength[k];
  const float radius = floorf(5.0f * s);
  const int pi = pos[2 * k + 0];
  const int pj = pos[2 * k + 1];
  const float inv = -0.5f / (4.0f * s * s);
  const float c0 = (*g_lratio) * s;
  for (int t = threadIdx.x; t < SIDE_C * SIDE_C; t += TPB) {
    const int di = t / SIDE_C - RMAX_C;
    const int dj = t % SIDE_C - RMAX_C;
    if (fabsf((float)di) <= radius && fabsf((float)dj) <= radius) {
      const int ni = pi + di, nj = pj + dj;
      if (ni >= 0 && ni < H_DIM && nj >= 0 && nj < W_DIM) {
        const float dist2 = (float)(di * di + dj * dj);
        const float c = expf(dist2 * inv) * c0;
        atomicAdd(&delta[(size_t)ni * W_DIM + nj], c);
      }
    }
  }
}

// ---------------------------------------------------------------------------
// 2) Fused elementwise: async-stage field/signal through LDS (double buffer),
//    read delta (in d_out), write field_update in place, emit block partials.
// ---------------------------------------------------------------------------
__global__ __launch_bounds__(TPB) void fuse_kernel(
    const float* __restrict__ field, const float* __restrict__ signal,
    float* delta_out,  // in: delta; out: unscaled field_update
    const float* __restrict__ g_infl, const float* __restrict__ g_gratio,
    float* __restrict__ partials) {
  __shared__ v4f sF[2][TPB];
  __shared__ v4f sS[2][TPB];
  __shared__ float redF[TPB / 32];
  __shared__ float redU[TPB / 32];

  const int tid = threadIdx.x;
  const unsigned blk = blockIdx.x;
  const float infl = *g_infl;
  const float grat = *g_gratio;

  const size_t base_elem = (size_t)blk * ELEMS_PER_BLOCK;
  const unsigned ldsF[2] = {lds_b32_addr(&sF[0][tid]), lds_b32_addr(&sF[1][tid])};
  const unsigned ldsS[2] = {lds_b32_addr(&sS[0][tid]), lds_b32_addr(&sS[1][tid])};
  const unsigned long long fbase = (unsigned long long)field;
  const unsigned long long sbase = (unsigned long long)signal;

  // Issue one 16B async global->LDS copy per lane for field and signal.
  auto issue = [&](int buf, int it) {
    const unsigned voff =
        (unsigned)((base_elem + (size_t)it * TPB * 4 + (size_t)tid * 4) *
                   sizeof(float));
    asm volatile("global_load_async_to_lds_b128 %0, %1, %2 th:TH_LOAD_NT"
                 :: "v"(ldsF[buf]), "v"(voff), "s"(fbase) : "memory");
    asm volatile("global_load_async_to_lds_b128 %0, %1, %2 th:TH_LOAD_NT"
                 :: "v"(ldsS[buf]), "v"(voff), "s"(sbase) : "memory");
  };

  issue(0, 0);

  const v4f* din4 = (const v4f*)delta_out;
  v4f* dout4 = (v4f*)delta_out;
  float accF = 0.0f, accU = 0.0f;

  for (int it = 0; it < ITER; ++it) {
    const int buf = it & 1;
    const size_t gidx4 = base_elem / 4 + (size_t)it * TPB + tid;
    const v4f d4 = din4[gidx4];  // delta: likely L2-resident (just written)

    if (it + 1 < ITER) {
      issue(buf ^ 1, it + 1);
      asm volatile("s_wait_asynccnt 0x2" ::: "memory");  // current chunk done
    } else {
      asm volatile("s_wait_asynccnt 0x0" ::: "memory");
    }

    const v4f f4 = sF[buf][tid];  // each lane reads only its own staged data
    const v4f s4 = sS[buf][tid];

    v4f u4;
    {
      float m;
      m = infl / (1.0f + expf(-(grat + d4.x)));
      u4.x = f4.x * (1.0f - m) + s4.x * m;
      m = infl / (1.0f + expf(-(grat + d4.y)));
      u4.y = f4.y * (1.0f - m) + s4.y * m;
      m = infl / (1.0f + expf(-(grat + d4.z)));
      u4.z = f4.z * (1.0f - m) + s4.z * m;
      m = infl / (1.0f + expf(-(grat + d4.w)));
      u4.w = f4.w * (1.0f - m) + s4.w * m;
    }
    dout4[gidx4] = u4;  // temporal store: re-read by scale pass, fits in L2

    accF += f4.x * f4.x + f4.y * f4.y + f4.z * f4.z + f4.w * f4.w;
    accU += u4.x * u4.x + u4.y * u4.y + u4.z * u4.z + u4.w * u4.w;
  }

  // wave32 reduction, then cross-wave via LDS (fixed shape -> deterministic)
  for (int off = 16; off > 0; off >>= 1) {
    accF += __shfl_xor(accF, off, 32);
    accU += __shfl_xor(accU, off, 32);
  }
  if ((tid & 31) == 0) {
    redF[tid >> 5] = accF;
    redU[tid >> 5] = accU;
  }
  __syncthreads();
  if (tid == 0) {
    float tF = 0.0f, tU = 0.0f;
    for (int i = 0; i < TPB / 32; ++i) { tF += redF[i]; tU += redU[i]; }
    partials[blk] = tF;
    partials[NBLK + blk] = tU;
  }
}

// ---------------------------------------------------------------------------
// 3) Finalize: deterministic fixed-order fp64 sum of partials -> scale factor.
// ---------------------------------------------------------------------------
__global__ __launch_bounds__(TPB) void finalize_kernel(float* wsf) {
  __shared__ double sF[TPB];
  __shared__ double sU[TPB];
  double aF = 0.0, aU = 0.0;
  for (int i = threadIdx.x; i < NBLK; i += TPB) {
    aF += (double)wsf[i];
    aU += (double)wsf[NBLK + i];
  }
  sF[threadIdx.x] = aF;
  sU[threadIdx.x] = aU;
  __syncthreads();
  for (int off = TPB / 2; off > 0; off >>= 1) {
    if (threadIdx.x < off) {
      sF[threadIdx.x] += sF[threadIdx.x + off];
      sU[threadIdx.x] += sU[threadIdx.x + off];
    }
    __syncthreads();
  }
  if (threadIdx.x == 0) {
    const double f = sF[0], u = sU[0];
    wsf[SCALE_SLOT] = (u > 0.0) ? (float)sqrt(f / u) : 1.0f;
  }
}

// ---------------------------------------------------------------------------
// 4) Rescale output in place; final stores are non-temporal (never re-read).
// ---------------------------------------------------------------------------
__global__ __launch_bounds__(TPB) void scale_kernel(float* out,
                                                    const float* __restrict__ wsf) {
  const float sc = wsf[SCALE_SLOT];
  const size_t i = (size_t)blockIdx.x * TPB + threadIdx.x;
  v4f* o4 = (v4f*)out;
  v4f v = o4[i];
  v.x *= sc; v.y *= sc; v.z *= sc; v.w *= sc;
  __builtin_nontemporal_store(v, &o4[i]);
}

// ---------------------------------------------------------------------------
extern "C" void kernel_launch(void* const* d_in, const int* in_sizes, int n_in,
                              void* d_out, int out_size, void* d_ws,
                              size_t ws_size, hipStream_t stream) {
  (void)in_sizes; (void)n_in; (void)out_size; (void)ws_size;
  const float* field    = (const float*)d_in[0];
  const float* signal   = (const float*)d_in[1];
  const int*   pos      = (const int*)d_in[2];
  const float* strength = (const float*)d_in[3];
  const float* infl     = (const float*)d_in[4];
  const float* gratio   = (const float*)d_in[5];
  const float* lratio   = (const float*)d_in[6];
  float* out = (float*)d_out;
  float* wsf = (float*)d_ws;

  // d_out serves as the zero-initialized delta accumulator, then is
  // overwritten in place with the (unscaled, then scaled) field update.
  (void)hipMemsetAsync(d_out, 0, NPIX * sizeof(float), stream);
  scatter_kernel<<<K_ATT, TPB, 0, stream>>>(pos, strength, lratio, out);
  fuse_kernel<<<NBLK, TPB, 0, stream>>>(field, signal, out, infl, gratio, wsf);
  finalize_kernel<<<1, TPB, 0, stream>>>(wsf);
  scale_kernel<<<(int)(NPIX / (4 * TPB)), TPB, 0, stream>>>(out, wsf);
}